// LearnableFuzzyAttention_49022756716874
// MI455X (gfx1250) — compile-verified
//
#include <hip/hip_runtime.h>
#include <hip/hip_bf16.h>
#include <math.h>

#define B 4
#define S 512
#define D 512
#define H 8
#define HD 64
#define F 5
#define EPSF 1e-8f

#define KC      32    // K-chunk staged in LDS
#define PSTRIDE 160   // dwords per k-pair row (128 data + 32 pad): half-offset = 160%64 = 32 banks

typedef __attribute__((ext_vector_type(2))) float v2f;
typedef __attribute__((ext_vector_type(8))) float v8f;

// ---------------------------------------------------------------------------
// Shared WMMA GEMM block routine (exact fp32 via V_WMMA_F32_16X16X4_F32).
//
// Block = 256 threads = 8 waves. Each wave owns a 16-row strip and the full
// 64-column width of the block tile (4 v8f accumulators -> 4 WMMAs / k-step,
// A fragment reused 4x).
//
// B chunks (KC x 64) are staged into LDS *k-pair interleaved*:
//   Bs[(k/2)*PSTRIDE + 2*col + (k&1)] = B[k][col]
// so each WMMA B-fragment {B[k][c], B[k+1][c]} is a single aligned
// ds_load_b64 straight into an even VGPR pair (no repack movs).
//
// Pointers are pre-offset to the block tile:
//   A    -> first row of this block's 128-row strip   (row stride lda)
//   Bsrc -> B[0, col0] of this block's 64-col strip   (row stride ldb)
//   bias -> bias[col0] or nullptr
//   C    -> C[row0, col0]                             (row stride ldc)
// ---------------------------------------------------------------------------
__device__ __forceinline__
void gemm_block_wmma(const float* __restrict__ A, int lda,
                     const float* __restrict__ Bsrc, int ldb,
                     const float* __restrict__ bias,
                     float* __restrict__ C, int ldc,
                     int Kdim)
{
    __shared__ __align__(16) float Bs[(KC / 2) * PSTRIDE];

    const int wave = threadIdx.x >> 5;
    const int lane = threadIdx.x & 31;
    const int half = lane >> 4;     // lanes 0-15 vs 16-31
    const int l16  = lane & 15;

    // A row owned by this lane (ISA: A 16x4 fp32 -> M = lane%16)
    const float* __restrict__ arow = A + (size_t)(wave * 16 + l16) * lda;

    // cooperative B-staging coordinates: thread t owns k-pair sp, cols sc..sc+3
    const int sp = threadIdx.x >> 4;         // 0..15 (k rows 2sp, 2sp+1)
    const int sc = (threadIdx.x & 15) << 2;  // 0,4,...,60

    const int col2 = l16 * 2;                // interleaved col offset (dwords)

    v8f acc0 = {}, acc1 = {}, acc2 = {}, acc3 = {};

    for (int k0 = 0; k0 < Kdim; k0 += KC) {
        // ---- stage B chunk (KC x 64) into LDS, b128-coalesced, pair-interleaved ----
        const float4 r0 = *(const float4*)(Bsrc + (size_t)(k0 + 2 * sp)     * ldb + sc);
        const float4 r1 = *(const float4*)(Bsrc + (size_t)(k0 + 2 * sp + 1) * ldb + sc);
        float* __restrict__ srow = &Bs[sp * PSTRIDE + 2 * sc];
        *(float4*)(srow + 0) = make_float4(r0.x, r1.x, r0.y, r1.y);
        *(float4*)(srow + 4) = make_float4(r0.z, r1.z, r0.w, r1.w);

        // prefetch next chunk (gfx1250 global_prefetch_b8)
        if (k0 + KC < Kdim)
            __builtin_prefetch(Bsrc + (size_t)(k0 + KC + 2 * sp) * ldb + sc, 0, 0);

        __syncthreads();

        #pragma unroll
        for (int kk = 0; kk < KC; kk += 4) {
            // A fragment: VGPR0 = K(0|2), VGPR1 = K(1|3) per lane-half (one b64)
            v2f af;
            af.x = arow[k0 + kk + half * 2 + 0];
            af.y = arow[k0 + kk + half * 2 + 1];

            // B fragments: pair row (kk/2 + half), one ds_load_b64 per col tile
            const float* __restrict__ brow = &Bs[((kk >> 1) + half) * PSTRIDE + col2];
            v2f b0 = *(const v2f*)(brow + 0);
            v2f b1 = *(const v2f*)(brow + 32);
            v2f b2 = *(const v2f*)(brow + 64);
            v2f b3 = *(const v2f*)(brow + 96);

            acc0 = __builtin_amdgcn_wmma_f32_16x16x4_f32(false, af, false, b0, (short)0, acc0, false, false);
            acc1 = __builtin_amdgcn_wmma_f32_16x16x4_f32(false, af, false, b1, (short)0, acc1, false, false);
            acc2 = __builtin_amdgcn_wmma_f32_16x16x4_f32(false, af, false, b2, (short)0, acc2, false, false);
            acc3 = __builtin_amdgcn_wmma_f32_16x16x4_f32(false, af, false, b3, (short)0, acc3, false, false);
        }
        __syncthreads();
    }

    // ---- epilogue: C/D layout VGPR r -> row (r + half*8), col = l16 ----
    float bb0 = 0.f, bb1 = 0.f, bb2 = 0.f, bb3 = 0.f;
    if (bias) {
        bb0 = bias[l16]; bb1 = bias[16 + l16]; bb2 = bias[32 + l16]; bb3 = bias[48 + l16];
    }
    float* __restrict__ cbase = C + (size_t)(wave * 16 + half * 8) * ldc;
    #pragma unroll
    for (int r = 0; r < 8; ++r) {
        float* __restrict__ cr = cbase + (size_t)r * ldc;
        cr[ 0 + l16] = acc0[r] + bb0;
        cr[16 + l16] = acc1[r] + bb1;
        cr[32 + l16] = acc2[r] + bb2;
        cr[48 + l16] = acc3[r] + bb3;
    }
}

// ---------------------------------------------------------------------------
// Projection GEMM: C(2048x512) = A(2048x512) @ W(512x512) + bias
// grid = (2048/128, 512/64) = (16, 8), block 256
// ---------------------------------------------------------------------------
__global__ __launch_bounds__(256)
void proj_gemm_wmma(const float* __restrict__ A, const float* __restrict__ W,
                    const float* __restrict__ bias, float* __restrict__ C)
{
    const int rowb = blockIdx.x * 128;
    const int colb = blockIdx.y * 64;
    gemm_block_wmma(A + (size_t)rowb * D, D,
                    W + colb, D,
                    bias + colb,
                    C + (size_t)rowb * D + colb, D,
                    D);
}

// ---------------------------------------------------------------------------
// attn @ V per (b,h): M=512, N=64, K=512; output in (B,S,D) layout.
// grid = (512/128, 1, B*H) = (4, 1, 32), block 256
// ---------------------------------------------------------------------------
__global__ __launch_bounds__(256)
void attnv_gemm_wmma(const float* __restrict__ attn, const float* __restrict__ v,
                     float* __restrict__ o_tmp)
{
    const int bh = blockIdx.z;
    const int b  = bh >> 3;
    const int h  = bh & 7;
    const int rowb = blockIdx.x * 128;
    gemm_block_wmma(attn + ((size_t)bh * S + rowb) * S, S,
                    v + (size_t)b * S * D + h * HD, D,
                    nullptr,
                    o_tmp + (size_t)(b * S + rowb) * D + h * HD, D,
                    S);
}

// ---------------------------------------------------------------------------
// Gaussian membership: qm/km[b,h,i,f] = exp(-0.5*sum_d nd^2) * gw[f]
// grid = (ceil(B*H*S*F/256), 2), blockIdx.y selects q vs k.
// ---------------------------------------------------------------------------
__global__ __launch_bounds__(256)
void membership_kernel(const float* __restrict__ q, const float* __restrict__ k,
                       const float* __restrict__ centers, const float* __restrict__ sigmas,
                       const float* __restrict__ gw,
                       float* __restrict__ qm, float* __restrict__ km)
{
    const int t = blockIdx.x * blockDim.x + threadIdx.x;
    if (t >= B * H * S * F) return;
    const float* __restrict__ src = blockIdx.y ? k : q;
    float* __restrict__ dst = blockIdx.y ? km : qm;

    const int f = t % F;
    const int i = (t / F) % S;
    const int h = (t / (F * S)) % H;
    const int b = t / (F * S * H);

    const float* __restrict__ zrow = src + ((size_t)(b * S + i)) * D + h * HD;
    const float* __restrict__ cen  = centers + f * HD;
    const float* __restrict__ sg   = sigmas + f * HD;

    float acc = 0.f;
    #pragma unroll 8
    for (int d = 0; d < HD; ++d) {
        float nd = (zrow[d] - cen[d]) / (sg[d] + EPSF);
        acc += nd * nd;
    }
    dst[t] = __expf(-0.5f * acc) * gw[f];
}

// ---------------------------------------------------------------------------
// Fuzzy scores + row softmax. One block (512 threads = 16 waves) per
// (b,h,i) row; thread j computes score(i,j) then block softmax over j.
// ---------------------------------------------------------------------------
__global__ __launch_bounds__(512)
void scores_softmax_kernel(const float* __restrict__ qm, const float* __restrict__ km,
                           const float* __restrict__ tnw, float* __restrict__ attn)
{
    __shared__ float red[16];
    __shared__ float qrow[F];
    __shared__ float w[3];

    const int rowid = blockIdx.x;        // (b*H + h)*S + i
    const int bh = rowid / S;
    const int j = threadIdx.x;
    const int wave = threadIdx.x >> 5;
    const int lane = threadIdx.x & 31;

    if (threadIdx.x < F) qrow[threadIdx.x] = qm[(size_t)rowid * F + threadIdx.x];
    if (threadIdx.x == 0) {
        float m = fmaxf(tnw[0], fmaxf(tnw[1], tnw[2]));
        float e0 = __expf(tnw[0] - m), e1 = __expf(tnw[1] - m), e2 = __expf(tnw[2] - m);
        float s = e0 + e1 + e2;
        w[0] = e0 / s; w[1] = e1 / s; w[2] = e2 / s;
    }
    __syncthreads();

    const float* __restrict__ krow = km + ((size_t)bh * S + j) * F;
    float prod = 0.f, mn = 0.f, lk = 0.f;
    #pragma unroll
    for (int f = 0; f < F; ++f) {
        float a = qrow[f], b = krow[f];
        prod += a * b;
        mn   += fminf(a, b);
        lk   += fmaxf(a + b - 1.f, 0.f);
    }
    float sc = w[0] * prod + w[1] * mn + w[2] * lk;

    // --- block max (wave32 shuffle + LDS tree) ---
    float m = sc;
    #pragma unroll
    for (int off = 16; off > 0; off >>= 1) m = fmaxf(m, __shfl_xor(m, off, 32));
    if (lane == 0) red[wave] = m;
    __syncthreads();
    if (wave == 0) {
        float t = (lane < 16) ? red[lane] : -3.402823e38f;
        #pragma unroll
        for (int off = 8; off > 0; off >>= 1) t = fmaxf(t, __shfl_xor(t, off, 32));
        if (lane == 0) red[0] = t;
    }
    __syncthreads();
    const float rowmax = red[0];
    const float e = __expf(sc - rowmax);
    __syncthreads();

    // --- block sum ---
    float ssum = e;
    #pragma unroll
    for (int off = 16; off > 0; off >>= 1) ssum += __shfl_xor(ssum, off, 32);
    if (lane == 0) red[wave] = ssum;
    __syncthreads();
    if (wave == 0) {
        float t = (lane < 16) ? red[lane] : 0.f;
        #pragma unroll
        for (int off = 8; off > 0; off >>= 1) t += __shfl_xor(t, off, 32);
        if (lane == 0) red[0] = t;
    }
    __syncthreads();

    attn[(size_t)rowid * S + j] = e / red[0];
}

// ---------------------------------------------------------------------------
extern "C" void kernel_launch(void* const* d_in, const int* in_sizes, int n_in,
                              void* d_out, int out_size, void* d_ws, size_t ws_size,
                              hipStream_t stream)
{
    const float* x       = (const float*)d_in[0];
    const float* Wq      = (const float*)d_in[1];
    const float* bq      = (const float*)d_in[2];
    const float* Wk      = (const float*)d_in[3];
    const float* bk      = (const float*)d_in[4];
    const float* Wv      = (const float*)d_in[5];
    const float* bv      = (const float*)d_in[6];
    const float* Wo      = (const float*)d_in[7];
    const float* bo      = (const float*)d_in[8];
    const float* centers = (const float*)d_in[9];
    const float* sigmas  = (const float*)d_in[10];
    const float* gw      = (const float*)d_in[11];
    const float* tnw     = (const float*)d_in[12];

    float* out  = (float*)d_out;                       // (B,S,D)
    float* attn = (float*)d_out + (size_t)B * S * D;   // (B,H,S,S)

    float* ws    = (float*)d_ws;
    float* q     = ws;
    float* k     = q  + (size_t)B * S * D;
    float* v     = k  + (size_t)B * S * D;
    float* qm    = v  + (size_t)B * S * D;
    float* km    = qm + (size_t)B * H * S * F;
    float* o_tmp = km + (size_t)B * H * S * F;

    const dim3 gemmGrid(B * S / 128, D / 64);
    const dim3 gemmBlk(256);

    // 1) Q/K/V projections (fp32 WMMA, LDS-staged B)
    proj_gemm_wmma<<<gemmGrid, gemmBlk, 0, stream>>>(x, Wq, bq, q);
    proj_gemm_wmma<<<gemmGrid, gemmBlk, 0, stream>>>(x, Wk, bk, k);
    proj_gemm_wmma<<<gemmGrid, gemmBlk, 0, stream>>>(x, Wv, bv, v);

    // 2) Gaussian memberships
    const int nm = B * H * S * F;
    membership_kernel<<<dim3((nm + 255) / 256, 2), 256, 0, stream>>>(
        q, k, centers, sigmas, gw, qm, km);

    // 3) Fuzzy scores + softmax -> attn (in d_out)
    scores_softmax_kernel<<<B * H * S, 512, 0, stream>>>(qm, km, tnw, attn);

    // 4) attn @ V (fp32 WMMA, LDS-staged V), written in (B,S,D) layout
    attnv_gemm_wmma<<<dim3(S / 128, 1, B * H), 256, 0, stream>>>(attn, v, o_tmp);

    // 5) Output projection (fp32 WMMA)
    proj_gemm_wmma<<<gemmGrid, gemmBlk, 0, stream>>>(o_tmp, Wo, bo, out);
}